// yoloLoss_78185584656879
// MI455X (gfx1250) — compile-verified
//
#include <hip/hip_runtime.h>

// ---------------------------------------------------------------------------
// YOLO loss on MI455X (gfx1250): pure streaming reduction, HBM-bound.
//   bytes = 257 MB @ 23.3 TB/s => ~11us floor; FLOPs negligible.
// Strategy: NT vector loads + global prefetch for the 1-pass stream,
// deterministic two-stage reduction; final 32-lane reduce uses
// V_WMMA_F32_16X16X4_F32 (ones x partials, exact fp32).
// ---------------------------------------------------------------------------

typedef __attribute__((ext_vector_type(2))) float v2f;
typedef __attribute__((ext_vector_type(8))) float v8f;

#define YL_S 14
#define YL_BLOCK 256
#define YL_NB 1568   // partials in d_ws: 1568 floats = 6.1 KB

#if defined(__has_builtin)
#if __has_builtin(__builtin_amdgcn_wmma_f32_16x16x4_f32)
#define YL_HAVE_WMMA_F32X4 1
#endif
#endif

__device__ __forceinline__ float yl_sq(float x) { return x * x; }

__global__ void __launch_bounds__(YL_BLOCK)
yolo_loss_partial(const float* __restrict__ pred,
                  const float* __restrict__ targ,
                  float* __restrict__ partials, int ncells) {
    const int tid = threadIdx.x;
    const int stride = gridDim.x * YL_BLOCK;
    const float inv14 = 1.0f / 14.0f;
    float acc = 0.0f;

    for (int cell = blockIdx.x * YL_BLOCK + tid; cell < ncells; cell += stride) {
        const v2f* pp = (const v2f*)(pred + (size_t)cell * 10);
        const v2f* tp = (const v2f*)(targ + (size_t)cell * 10);

        // Prefetch next grid-stride iteration of this thread's stream.
        __builtin_prefetch(pp + (size_t)stride * 5, 0, 0);
        __builtin_prefetch(tp + (size_t)stride * 5, 0, 0);

        // Non-temporal: stream-once data, keep it out of L2 residency.
        v2f p01 = __builtin_nontemporal_load(pp + 0);
        v2f p23 = __builtin_nontemporal_load(pp + 1);
        v2f p45 = __builtin_nontemporal_load(pp + 2);
        v2f p67 = __builtin_nontemporal_load(pp + 3);
        v2f p89 = __builtin_nontemporal_load(pp + 4);
        v2f t01 = __builtin_nontemporal_load(tp + 0);
        v2f t23 = __builtin_nontemporal_load(tp + 1);
        v2f t45 = __builtin_nontemporal_load(tp + 2);
        v2f t89 = __builtin_nontemporal_load(tp + 4);  // t[5..8] unused by the loss

        const float t0 = t01.x, t1 = t01.y, t2 = t23.x, t3 = t23.y;
        const float t4 = t45.x, t9 = t89.y;

        const float objf = (t4 > 0.0f) ? 1.0f : 0.0f;
        const float noo  = (1.0f - objf) * (yl_sq(p45.x - t4) + yl_sq(p89.y - t9));

        // Box 0: (x,y,w,h,c) = (p0,p1,p2,p3,p4)
        // gab = (pw+tw)/2 - |px - tx|/14, clamped to [0, min(pw,tw)]
        float g0x = 0.5f * (p23.x + t2) - fabsf(p01.x - t0) * inv14;
        float g0y = 0.5f * (p23.y + t3) - fabsf(p01.y - t1) * inv14;
        g0x = fminf(fminf(fmaxf(g0x, 0.0f), p23.x), t2);
        g0y = fminf(fminf(fmaxf(g0y, 0.0f), p23.y), t3);
        const float iou0 = (g0x * g0y) / (p23.x * p23.y);

        // Box 1: (x,y,w,h,c) = (p5,p6,p7,p8,p9)
        float g1x = 0.5f * (p67.y + t2) - fabsf(p45.y - t0) * inv14;
        float g1y = 0.5f * (p89.x + t3) - fabsf(p67.x - t1) * inv14;
        g1x = fminf(fminf(fmaxf(g1x, 0.0f), p67.y), t2);
        g1y = fminf(fminf(fmaxf(g1y, 0.0f), p89.x), t3);
        const float iou1 = (g1x * g1y) / (p67.y * p89.x);

        // argmax with first-occurrence tie-break (box 0 wins ties)
        const bool sel = iou1 > iou0;
        const float rx = sel ? p45.y : p01.x;
        const float ry = sel ? p67.x : p01.y;
        const float rw = sel ? p67.y : p23.x;
        const float rh = sel ? p89.x : p23.y;
        const float rc = sel ? p89.y : p45.x;

        const float contain = yl_sq(rc - t4);
        const float xy = yl_sq(rx - t0) + yl_sq(ry - t1);
        const float wh = yl_sq(sqrtf(rw) - sqrtf(t2)) + yl_sq(sqrtf(rh) - sqrtf(t3));

        acc += objf * (5.0f * (xy + wh) + contain) + 0.5f * noo;
    }

    // wave32 reduction (fixed order -> deterministic)
    acc += __shfl_xor(acc, 16, 32);
    acc += __shfl_xor(acc, 8, 32);
    acc += __shfl_xor(acc, 4, 32);
    acc += __shfl_xor(acc, 2, 32);
    acc += __shfl_xor(acc, 1, 32);

    __shared__ float wsum[YL_BLOCK / 32];
    const int lane = tid & 31, wave = tid >> 5;
    if (lane == 0) wsum[wave] = acc;
    __syncthreads();
    if (tid == 0) {
        float s = 0.0f;
        for (int w = 0; w < YL_BLOCK / 32; ++w) s += wsum[w];
        partials[blockIdx.x] = s;   // written unconditionally by every block
    }
}

__global__ void __launch_bounds__(YL_BLOCK)
yolo_final_reduce(const float* __restrict__ partials, float* __restrict__ out,
                  int nparts, float invN) {
    __shared__ float lds[YL_BLOCK];
    const int tid = threadIdx.x;

    float s = 0.0f;
    for (int i = tid; i < nparts; i += YL_BLOCK) s += partials[i];
    lds[tid] = s;
    __syncthreads();

    float v = 0.0f;
    if (tid < 32) {
        for (int i = tid; i < YL_BLOCK; i += 32) v += lds[i];
    }
    // Control flow reconverged here: EXEC is all-1s in every wave (WMMA req).
    // Only wave 0 carries data; other waves feed zeros (harmless).
#ifdef YL_HAVE_WMMA_F32X4
    // D = ones(16x4) x B(4x16) + 0: D[m,n] = sum_k B[k,n].
    // Sum of D row 0 over n = sum of ALL 64 B entries = sum of the 32 v's,
    // exact in fp32. D row 0 lives in VGPR0 of lanes 0..15 (lanes 16..31
    // hold row M=8, an identical copy — we only reduce the low half).
    v2f a; a.x = 1.0f; a.y = 1.0f;
    v2f b; b.x = v;    b.y = 0.0f;
    v8f c = {};
    v8f d = __builtin_amdgcn_wmma_f32_16x16x4_f32(
        /*neg_a=*/false, a, /*neg_b=*/false, b,
        /*c_mod=*/(short)0, c, /*reuse_a=*/false, /*reuse_b=*/false);
    float r = d[0];
    r += __shfl_xor(r, 1, 32);
    r += __shfl_xor(r, 2, 32);
    r += __shfl_xor(r, 4, 32);
    r += __shfl_xor(r, 8, 32);   // lane 0: total over lanes 0..15 = full sum
#else
    float r = v;
    r += __shfl_xor(r, 16, 32);
    r += __shfl_xor(r, 8, 32);
    r += __shfl_xor(r, 4, 32);
    r += __shfl_xor(r, 2, 32);
    r += __shfl_xor(r, 1, 32);
#endif
    if (tid == 0) out[0] = r * invN;
}

extern "C" void kernel_launch(void* const* d_in, const int* in_sizes, int n_in,
                              void* d_out, int out_size, void* d_ws, size_t ws_size,
                              hipStream_t stream) {
    const float* pred = (const float*)d_in[0];
    const float* targ = (const float*)d_in[1];
    float* out = (float*)d_out;
    float* partials = (float*)d_ws;   // YL_NB floats

    const int total  = in_sizes[0];               // N * 14 * 14 * 10
    const int ncells = total / 10;
    const int N      = total / (YL_S * YL_S * 10);

    yolo_loss_partial<<<YL_NB, YL_BLOCK, 0, stream>>>(pred, targ, partials, ncells);
    yolo_final_reduce<<<1, YL_BLOCK, 0, stream>>>(partials, out, YL_NB, 1.0f / (float)N);
}